// GNNLayer_44495861187321
// MI455X (gfx1250) — compile-verified
//
#include <hip/hip_runtime.h>
#include <math.h>

// Problem constants (fixed by the reference).
#define UD  256
#define ND  50000
#define ED  320000
#define EPS 1e-5f

// Padded LDS row stride for the 16x256 A tile: 260 floats = 1040 B.
// (16B aligned for float4 stores; fragment reads spread over all 64 banks.)
#define ASTRIDE 260

typedef float v2f __attribute__((ext_vector_type(2)));
typedef float v8f __attribute__((ext_vector_type(8)));

// Exact-fp32 matrix core op: D(16x16) = A(16x4) * B(4x16) + C, wave32.
__device__ __forceinline__ v8f wmma_f32_k4(v2f a, v2f b, v8f c) {
  return __builtin_amdgcn_wmma_f32_16x16x4_f32(false, a, false, b, (short)0, c,
                                               false, false);
}

__device__ __forceinline__ float sigmoidf_fast(float x) {
  return 1.0f / (1.0f + __expf(-x));
}
__device__ __forceinline__ float siluf(float x) {
  return x * sigmoidf_fast(x);
}

// Stage one 16-row x 256-col fp32 A tile into LDS (128 threads).
__device__ __forceinline__ void stage_a_tile(float* As, const float* A,
                                             int row0, int tid) {
  // 16*64 float4s, 8 per thread.
#pragma unroll
  for (int i = tid; i < 16 * 64; i += 128) {
    const int m  = i >> 6;
    const int c4 = i & 63;
    const float4 v = ((const float4*)(A + (size_t)(row0 + m) * UD))[c4];
    *(float4*)&As[m * ASTRIDE + c4 * 4] = v;
  }
  __syncthreads();
}

// -------------------------------------------------------------------------
// K0: zero a float region (scatter sums, counts, BN accumulators).
// -------------------------------------------------------------------------
__global__ __launch_bounds__(256) void zero_kernel(float* __restrict__ p, size_t n) {
  size_t i = (size_t)blockIdx.x * blockDim.x + threadIdx.x;
  if (i < n) p[i] = 0.0f;
}

// -------------------------------------------------------------------------
// K1: node GEMMs.  out_w = x @ w^T + b  for w1..w4.
// grid = N/16 blocks, block = 128 (4 waves).  The block stages the 16x256
// x-tile in LDS once (x read exactly once from HBM); wave w computes the
// whole 16x256 output of weight w in two 128-col passes with 8 accumulators,
// so 8 WMMAs share each A fragment (ds_load_b64) and B fragments stream from
// the L2-resident 256KB weight.
// -------------------------------------------------------------------------
__global__ __launch_bounds__(128) void node_gemm_kernel(
    const float* __restrict__ x,
    const float* __restrict__ w1, const float* __restrict__ b1,
    const float* __restrict__ w2, const float* __restrict__ b2,
    const float* __restrict__ w3, const float* __restrict__ b3,
    const float* __restrict__ w4, const float* __restrict__ b4,
    float* __restrict__ hv, float* __restrict__ x2o,
    float* __restrict__ x3o, float* __restrict__ x4o)
{
  __shared__ float As[16 * ASTRIDE];

  const int tid  = threadIdx.x;
  const int wave = tid >> 5;
  const int lane = tid & 31;
  const int row0 = blockIdx.x * 16;

  stage_a_tile(As, x, row0, tid);

  const float* W; const float* Bv; float* O;
  if (wave == 0)      { W = w1; Bv = b1; O = hv;  }
  else if (wave == 1) { W = w2; Bv = b2; O = x2o; }
  else if (wave == 2) { W = w3; Bv = b3; O = x3o; }
  else                { W = w4; Bv = b4; O = x4o; }

  const int mn = lane & 15;
  const int kh = (lane >> 4) << 1;   // 0 or 2
  const int mh = (lane >> 4) << 3;   // 0 or 8
  const float* as = &As[mn * ASTRIDE + kh];

#pragma unroll
  for (int g = 0; g < 2; ++g) {
    const int colbase = g * 128;
    const float* brow = W + (size_t)(colbase + mn) * UD + kh;

    v8f acc[8];
#pragma unroll
    for (int t = 0; t < 8; ++t) acc[t] = (v8f){0.f,0.f,0.f,0.f,0.f,0.f,0.f,0.f};

#pragma unroll 2
    for (int k = 0; k < UD; k += 4) {
      const v2f a = *(const v2f*)(as + k);
#pragma unroll
      for (int t = 0; t < 8; ++t) {
        const v2f b = *(const v2f*)(brow + (size_t)t * 16 * UD + k);
        acc[t] = wmma_f32_k4(a, b, acc[t]);
      }
    }

#pragma unroll
    for (int t = 0; t < 8; ++t) {
      const int c0 = colbase + t * 16;
      const float bias = Bv[c0 + mn];
#pragma unroll
      for (int v = 0; v < 8; ++v) {
        O[(size_t)(row0 + v + mh) * UD + c0 + mn] = acc[t][v] + bias;
      }
    }
  }
}

// -------------------------------------------------------------------------
// K2: edge GEMM.  he = edge_attr @ we^T + be.
// grid = E/16 blocks, block = 128 (4 waves).  Block stages the 16x256
// edge_attr tile once (328MB streamed exactly once); wave w computes cols
// [64w, 64w+64) with 4 accumulators (4 WMMAs share each A fragment).
// Light VGPR use -> high occupancy for the streaming-dominated GEMM.
// -------------------------------------------------------------------------
__global__ __launch_bounds__(128) void edge_gemm_kernel(
    const float* __restrict__ ea,
    const float* __restrict__ we, const float* __restrict__ be,
    float* __restrict__ he)
{
  __shared__ float As[16 * ASTRIDE];

  const int tid  = threadIdx.x;
  const int wave = tid >> 5;
  const int lane = tid & 31;
  const int row0 = blockIdx.x * 16;

  stage_a_tile(As, ea, row0, tid);

  const int mn = lane & 15;
  const int kh = (lane >> 4) << 1;
  const int mh = (lane >> 4) << 3;
  const float* as = &As[mn * ASTRIDE + kh];

  const int col0 = wave * 64;
  const float* brow = we + (size_t)(col0 + mn) * UD + kh;

  v8f acc0 = {0.f,0.f,0.f,0.f,0.f,0.f,0.f,0.f};
  v8f acc1 = acc0, acc2 = acc0, acc3 = acc0;

#pragma unroll 4
  for (int k = 0; k < UD; k += 4) {
    const v2f a  = *(const v2f*)(as + k);
    const v2f b0 = *(const v2f*)(brow + k);
    const v2f b1 = *(const v2f*)(brow + (size_t)16 * UD + k);
    const v2f b2 = *(const v2f*)(brow + (size_t)32 * UD + k);
    const v2f b3 = *(const v2f*)(brow + (size_t)48 * UD + k);
    acc0 = wmma_f32_k4(a, b0, acc0);
    acc1 = wmma_f32_k4(a, b1, acc1);
    acc2 = wmma_f32_k4(a, b2, acc2);
    acc3 = wmma_f32_k4(a, b3, acc3);
  }

  v8f accs[4] = {acc0, acc1, acc2, acc3};
#pragma unroll
  for (int t = 0; t < 4; ++t) {
    const int c0 = col0 + t * 16;
    const float bias = be[c0 + mn];
#pragma unroll
    for (int v = 0; v < 8; ++v) {
      he[(size_t)(row0 + v + mh) * UD + c0 + mn] = accs[t][v] + bias;
    }
  }
}

// -------------------------------------------------------------------------
// K3: gather-gate-scatter.  msg = sigmoid(edge_attr[e]) * x2[dst[e]];
// sums[src[e]] += msg (f32 atomics, L2-resident 51MB target); cnt[src[e]] += 1.
// -------------------------------------------------------------------------
__global__ __launch_bounds__(256) void edge_aggregate_kernel(
    const float* __restrict__ ea, const float* __restrict__ x2,
    const int* __restrict__ src, const int* __restrict__ dst,
    float* __restrict__ sums, float* __restrict__ cnt)
{
  const int e = blockIdx.x * 8 + (threadIdx.x >> 5);
  if (e >= ED) return;
  const int lane = threadIdx.x & 31;
  const int s = src[e];
  const int d = dst[e];

  const float4* earow = (const float4*)(ea + (size_t)e * UD);
  const float4* xrow  = (const float4*)(x2 + (size_t)d * UD);
  float* srow = sums + (size_t)s * UD;

#pragma unroll
  for (int c = lane; c < UD / 4; c += 32) {
    float4 ev = earow[c];
    float4 xv = xrow[c];
    atomicAdd(&srow[4 * c + 0], sigmoidf_fast(ev.x) * xv.x);
    atomicAdd(&srow[4 * c + 1], sigmoidf_fast(ev.y) * xv.y);
    atomicAdd(&srow[4 * c + 2], sigmoidf_fast(ev.z) * xv.z);
    atomicAdd(&srow[4 * c + 3], sigmoidf_fast(ev.w) * xv.w);
  }
  if (lane == 0) atomicAdd(&cnt[s], 1.0f);
}

// -------------------------------------------------------------------------
// K4: hv = x1 + sums/max(cnt,1); accumulate column sum/sumsq for node BN.
// -------------------------------------------------------------------------
__global__ __launch_bounds__(256) void node_h_stats_kernel(
    float* __restrict__ hv, const float* __restrict__ sums,
    const float* __restrict__ cnt,
    float* __restrict__ colsum, float* __restrict__ colsq, int rowsPerBlock)
{
  const int u = threadIdx.x;
  const int r0 = blockIdx.x * rowsPerBlock;
  int r1 = r0 + rowsPerBlock; if (r1 > ND) r1 = ND;

  float s = 0.0f, sq = 0.0f;
  for (int r = r0; r < r1; ++r) {
    const float inv = 1.0f / fmaxf(cnt[r], 1.0f);
    const size_t i = (size_t)r * UD + u;
    const float h = hv[i] + sums[i] * inv;
    hv[i] = h;
    s += h; sq += h * h;
  }
  atomicAdd(&colsum[u], s);
  atomicAdd(&colsq[u], sq);
}

// -------------------------------------------------------------------------
// K5: he += x3[src] + x4[dst]; accumulate column sum/sumsq for edge BN.
// -------------------------------------------------------------------------
__global__ __launch_bounds__(256) void edge_h_stats_kernel(
    float* __restrict__ he, const float* __restrict__ x3,
    const float* __restrict__ x4,
    const int* __restrict__ src, const int* __restrict__ dst,
    float* __restrict__ colsum, float* __restrict__ colsq, int edgesPerBlock)
{
  const int u = threadIdx.x;
  const int e0 = blockIdx.x * edgesPerBlock;
  int e1 = e0 + edgesPerBlock; if (e1 > ED) e1 = ED;

  float s = 0.0f, sq = 0.0f;
  for (int e = e0; e < e1; ++e) {
    const int sr = src[e];
    const int ds = dst[e];
    const size_t i = (size_t)e * UD + u;
    const float h = he[i] + x3[(size_t)sr * UD + u] + x4[(size_t)ds * UD + u];
    he[i] = h;
    s += h; sq += h * h;
  }
  atomicAdd(&colsum[u], s);
  atomicAdd(&colsq[u], sq);
}

// -------------------------------------------------------------------------
// K6/K7: finalize  out = input + silu( g*(h-m)/sqrt(v+eps) + b ).
// Biased batch stats: m = colsum/R, v = colsq/R - m*m.
// -------------------------------------------------------------------------
__global__ __launch_bounds__(256) void bn_silu_residual_kernel(
    const float* __restrict__ inp, const float* __restrict__ h,
    const float* __restrict__ colsum, const float* __restrict__ colsq,
    const float* __restrict__ g, const float* __restrict__ bt,
    float* __restrict__ out, long long rows)
{
  const long long idx = (long long)blockIdx.x * blockDim.x + threadIdx.x;
  if (idx >= rows * UD) return;
  const int u = (int)(idx & (UD - 1));
  const float invR = 1.0f / (float)rows;
  const float m = colsum[u] * invR;
  const float var = colsq[u] * invR - m * m;
  const float istd = rsqrtf(var + EPS);
  const float t = g[u] * (h[idx] - m) * istd + bt[u];
  out[idx] = inp[idx] + siluf(t);
}

// -------------------------------------------------------------------------
// Host launcher.
// -------------------------------------------------------------------------
extern "C" void kernel_launch(void* const* d_in, const int* in_sizes, int n_in,
                              void* d_out, int out_size, void* d_ws, size_t ws_size,
                              hipStream_t stream) {
  (void)in_sizes; (void)n_in; (void)out_size; (void)ws_size;

  const float* x   = (const float*)d_in[0];
  const int*   ei  = (const int*)d_in[1];      // [2, E]: row0=src, row1=dst
  const float* ea  = (const float*)d_in[2];
  const float* w1  = (const float*)d_in[3];  const float* b1 = (const float*)d_in[4];
  const float* w2  = (const float*)d_in[5];  const float* b2 = (const float*)d_in[6];
  const float* w3  = (const float*)d_in[7];  const float* b3 = (const float*)d_in[8];
  const float* w4  = (const float*)d_in[9];  const float* b4 = (const float*)d_in[10];
  const float* we  = (const float*)d_in[11]; const float* be = (const float*)d_in[12];
  const float* gv  = (const float*)d_in[13]; const float* btv = (const float*)d_in[14];
  const float* ge  = (const float*)d_in[15]; const float* bte = (const float*)d_in[16];

  const int* src = ei;
  const int* dst = ei + ED;

  const size_t NU = (size_t)ND * UD;
  const size_t EU = (size_t)ED * UD;

  // Workspace layout (floats), ~585 MB total.
  float* ws   = (float*)d_ws;
  float* hv   = ws;             // [N,U]  x1 (+agg later)
  float* x2   = hv   + NU;      // [N,U]
  float* x3   = x2   + NU;      // [N,U]
  float* x4   = x3   + NU;      // [N,U]
  float* he   = x4   + NU;      // [E,U]  w1e (+gathers later)
  float* sums = he   + EU;      // [N,U]  scatter target
  float* cnt  = sums + NU;      // [N]
  float* st   = cnt  + ND;      // 4*256 BN accumulators
  float* csum_v = st;           // [256]
  float* csq_v  = st + UD;      // [256]
  float* csum_e = st + 2 * UD;  // [256]
  float* csq_e  = st + 3 * UD;  // [256]

  float* out_x = (float*)d_out;       // [N,U]
  float* out_w = out_x + NU;          // [E,U]

  // K0: zero sums + cnt + stats (contiguous region).
  {
    const size_t nz = NU + (size_t)ND + 4 * UD;
    zero_kernel<<<(unsigned)((nz + 255) / 256), 256, 0, stream>>>(sums, nz);
  }

  // K1: node GEMMs (fp32 WMMA, LDS-staged A, 8-way register blocking).
  node_gemm_kernel<<<ND / 16, 128, 0, stream>>>(x, w1, b1, w2, b2, w3, b3,
                                                w4, b4, hv, x2, x3, x4);

  // K2: edge GEMM (fp32 WMMA, LDS-staged A, 4-way register blocking).
  edge_gemm_kernel<<<ED / 16, 128, 0, stream>>>(ea, we, be, he);

  // K3: gather-gate-scatter-mean numerators.
  edge_aggregate_kernel<<<ED / 8, 256, 0, stream>>>(ea, x2, src, dst, sums, cnt);

  // K4: node h = x1 + agg, column stats.
  {
    const int rpb = 100;                 // 500 blocks
    node_h_stats_kernel<<<(ND + rpb - 1) / rpb, 256, 0, stream>>>(
        hv, sums, cnt, csum_v, csq_v, rpb);
  }

  // K5: edge h = w1e + x3[src] + x4[dst], column stats.
  {
    const int epb = 160;                 // 2000 blocks
    edge_h_stats_kernel<<<(ED + epb - 1) / epb, 256, 0, stream>>>(
        he, x3, x4, src, dst, csum_e, csq_e, epb);
  }

  // K6: x_out = x + silu(BN(h_v)).
  bn_silu_residual_kernel<<<(unsigned)(NU / 256), 256, 0, stream>>>(
      x, hv, csum_v, csq_v, gv, btv, out_x, (long long)ND);

  // K7: w_out = edge_attr + silu(BN(h_e)).
  bn_silu_residual_kernel<<<(unsigned)(EU / 256), 256, 0, stream>>>(
      ea, he, csum_e, csq_e, ge, bte, out_w, (long long)ED);
}